// NKIExpertRouter_24970939859024
// MI455X (gfx1250) — compile-verified
//
#include <hip/hip_runtime.h>
#include <math.h>

typedef __attribute__((ext_vector_type(2))) float v2f;
typedef __attribute__((ext_vector_type(8))) float v8f;
typedef __attribute__((ext_vector_type(4))) unsigned int u32x4;
typedef __attribute__((ext_vector_type(8))) int i32x8;
typedef __attribute__((ext_vector_type(4))) int i32x4;

#define HIDDEN          4096
#define NEXP            64
#define TOPK            8
#define WAVES_PER_BLOCK 4
#define TOKENS_PER_WAVE 16
#define LDS_ROW         65            // padded logits row (floats)
#define KC              128           // K-chunk per TDM tile
#define NCHUNK          (HIDDEN / KC) // 32
#define NBUF            3             // triple buffer
#define ARS             (KC + 4)      // padded A-tile row stride (floats): banks = (4m+k)%64, conflict-free
#define TILE_FLOATS     (TOKENS_PER_WAVE * ARS)

#define HAVE_TDM __has_builtin(__builtin_amdgcn_tensor_load_to_lds)

#if HAVE_TDM
// One TDM descriptor: 2D tile, 16 rows x 128 f32, row stride 4096 f32 in memory,
// LDS-padded to 132 f32 per row (pad 4 DWORDs every 128 DWORDs).
__device__ __forceinline__ void tdm_load_tile(const float* gaddr, unsigned int lds_off) {
    const unsigned long long ga = (unsigned long long)(uintptr_t)gaddr;
    u32x4 g0;
    g0[0] = 1u;                                    // count=1 (valid user D#)
    g0[1] = lds_off;                               // LDS byte address
    g0[2] = (unsigned int)ga;                      // global_addr[31:0]
    g0[3] = (unsigned int)(ga >> 32) | (2u << 30); // global_addr[56:32] | type=2

    i32x8 g1;
    g1[0] = (int)((2u << 16)    // data_size = 4 bytes
                | (1u << 20)    // pad_enable
                | (6u << 22)    // pad_interval: 2^(6+1) = 128 DWORDs
                | (3u << 25));  // pad_amount:   3+1     = 4 DWORDs
    g1[1] = (int)(4096u << 16);            // tensor_dim0 lo16 -> bits[63:48]
    g1[2] = (int)(16u << 16);              // tensor_dim0 hi=0 | tensor_dim1=16
    g1[3] = (int)(128u << 16);             // tensor_dim1 hi=0 | tile_dim0=128
    g1[4] = (int)16u;                      // tile_dim1=16 | tile_dim2=0
    g1[5] = (int)4096u;                    // tensor_dim0_stride lo32
    g1[6] = 0;                             // stride0 hi | stride1 lo
    g1[7] = 0;                             // stride1 hi

    const i32x4 gz4 = {0, 0, 0, 0};
    const i32x8 gz8 = {0, 0, 0, 0, 0, 0, 0, 0};
    __builtin_amdgcn_tensor_load_to_lds(g0, g1, gz4, gz4, gz8, 0);
}
#endif

__global__ __launch_bounds__(WAVES_PER_BLOCK * 32)
void router_topk_kernel(const float* __restrict__ hidden,
                        const float* __restrict__ Wr,
                        float* __restrict__ out_w,
                        float* __restrict__ out_idx)
{
    __shared__ float atile[WAVES_PER_BLOCK][NBUF][TILE_FLOATS];
    __shared__ float logits[WAVES_PER_BLOCK][TOKENS_PER_WAVE][LDS_ROW];

    const int lane        = threadIdx.x & 31;
    const int waveInBlock = threadIdx.x >> 5;
    const int wave        = blockIdx.x * WAVES_PER_BLOCK + waveInBlock;
    const int row_base    = wave * TOKENS_PER_WAVE;

    // WMMA f32 16x16x4 operand mapping (wave32):
    //   A (16x4): lane l holds row M = l&15, K-pair base = 2*(l>>4)
    //   B (4x16): lane l holds col N = l&15, K-pair base = 2*(l>>4)
    const int m  = lane & 15;
    const int kh = (lane >> 4) << 1;

    const float* __restrict__ wbase = Wr + (size_t)m * HIDDEN + kh;

    v8f acc0 = {}; v8f acc1 = {}; v8f acc2 = {}; v8f acc3 = {};

#if HAVE_TDM
    const float* hid_tile = hidden + (size_t)row_base * HIDDEN; // wave-uniform
    const unsigned int lds_base  = (unsigned int)(uintptr_t)&atile[waveInBlock][0][0];
    const unsigned int tile_bytes = TILE_FLOATS * 4;

    // Consume one K-chunk out of LDS buffer `buf`, accumulating 4 WMMA tiles.
    auto compute_chunk = [&](int c, int buf) {
        const float* ap = &atile[waveInBlock][0][0] + buf * TILE_FLOATS + m * ARS + kh;
        const float* wp = wbase + c * KC;
#pragma unroll 4
        for (int kk = 0; kk < KC; kk += 4) {
            const v2f a  = *(const v2f*)(ap + kk);
            const v2f w0 = *(const v2f*)(wp + kk);
            const v2f w1 = *(const v2f*)(wp + 16 * HIDDEN + kk);
            const v2f w2 = *(const v2f*)(wp + 32 * HIDDEN + kk);
            const v2f w3 = *(const v2f*)(wp + 48 * HIDDEN + kk);
            acc0 = __builtin_amdgcn_wmma_f32_16x16x4_f32(false, a, false, w0, (short)0, acc0, false, false);
            acc1 = __builtin_amdgcn_wmma_f32_16x16x4_f32(false, a, false, w1, (short)0, acc1, false, false);
            acc2 = __builtin_amdgcn_wmma_f32_16x16x4_f32(false, a, false, w2, (short)0, acc2, false, false);
            acc3 = __builtin_amdgcn_wmma_f32_16x16x4_f32(false, a, false, w3, (short)0, acc3, false, false);
        }
    };
    auto issue_chunk = [&](int c) {
        tdm_load_tile(hid_tile + (size_t)c * KC, lds_base + (unsigned)(c % NBUF) * tile_bytes);
    };

    issue_chunk(0); issue_chunk(1); issue_chunk(2);
#pragma unroll 1
    for (int c = 0; c + 2 < NCHUNK; ++c) {
        __builtin_amdgcn_s_wait_tensorcnt(2);       // oldest (chunk c) has landed
        __asm__ volatile("" ::: "memory");
        compute_chunk(c, c % NBUF);
        if (c + 3 < NCHUNK) issue_chunk(c + 3);     // refill freed buffer
    }
    __builtin_amdgcn_s_wait_tensorcnt(1);
    __asm__ volatile("" ::: "memory");
    compute_chunk(NCHUNK - 2, (NCHUNK - 2) % NBUF);
    __builtin_amdgcn_s_wait_tensorcnt(0);
    __asm__ volatile("" ::: "memory");
    compute_chunk(NCHUNK - 1, (NCHUNK - 1) % NBUF);
#else
    // Fallback: direct global A loads.
    const float* aptr = hidden + (size_t)(row_base + m) * HIDDEN + kh;
#pragma unroll 4
    for (int k = 0; k < HIDDEN; k += 4) {
        const v2f a  = *(const v2f*)(aptr + k);
        const v2f w0 = *(const v2f*)(wbase + k);
        const v2f w1 = *(const v2f*)(wbase + 16 * HIDDEN + k);
        const v2f w2 = *(const v2f*)(wbase + 32 * HIDDEN + k);
        const v2f w3 = *(const v2f*)(wbase + 48 * HIDDEN + k);
        acc0 = __builtin_amdgcn_wmma_f32_16x16x4_f32(false, a, false, w0, (short)0, acc0, false, false);
        acc1 = __builtin_amdgcn_wmma_f32_16x16x4_f32(false, a, false, w1, (short)0, acc1, false, false);
        acc2 = __builtin_amdgcn_wmma_f32_16x16x4_f32(false, a, false, w2, (short)0, acc2, false, false);
        acc3 = __builtin_amdgcn_wmma_f32_16x16x4_f32(false, a, false, w3, (short)0, acc3, false, false);
    }
#endif

    // Scatter logits tile to LDS.
    // C layout: element (vgpr v, lane l): row M = v + 8*(l>>4), col N = (l&15) + 16*nt
    const int mhi = (lane >> 4) << 3;
#pragma unroll
    for (int v = 0; v < 8; ++v) {
        const int M = v + mhi;
        logits[waveInBlock][M][m +  0] = acc0[v];
        logits[waveInBlock][M][m + 16] = acc1[v];
        logits[waveInBlock][M][m + 32] = acc2[v];
        logits[waveInBlock][M][m + 48] = acc3[v];
    }
    __syncthreads();

    // Lanes 0..15: one token each. Iterative argmax top-8 (first-occurrence
    // tie-break == lax.top_k), then softmax over the selected 8 logits
    // (== renormalized top-k of the full softmax).
    if (lane < 16) {
        float* rowp = &logits[waveInBlock][lane][0];
        float selv[TOPK];
        int   seli[TOPK];
#pragma unroll
        for (int t = 0; t < TOPK; ++t) {
            float best = -INFINITY;
            int   bi   = 0;
            for (int e = 0; e < NEXP; ++e) {
                const float x = rowp[e];
                if (x > best) { best = x; bi = e; }
            }
            selv[t] = best;
            seli[t] = bi;
            rowp[bi] = -INFINITY;
        }
        const float m0 = selv[0];
        float w[TOPK];
        float s = 0.0f;
#pragma unroll
        for (int t = 0; t < TOPK; ++t) { w[t] = __expf(selv[t] - m0); s += w[t]; }
        const float inv = 1.0f / s;

        const size_t tok = (size_t)row_base + lane;
        float* ow = out_w   + tok * TOPK;
        float* oi = out_idx + tok * TOPK;
#pragma unroll
        for (int t = 0; t < TOPK; ++t) {
            ow[t] = w[t] * inv;
            oi[t] = (float)seli[t];
        }
    }
}

extern "C" void kernel_launch(void* const* d_in, const int* in_sizes, int n_in,
                              void* d_out, int out_size, void* d_ws, size_t ws_size,
                              hipStream_t stream) {
    const float* hidden = (const float*)d_in[0];   // [4,4096,4096] f32
    const float* Wr     = (const float*)d_in[1];   // [64,4096]     f32

    const int ntok = in_sizes[0] / HIDDEN;         // 16384 tokens
    float* out_w   = (float*)d_out;                        // [ntok,8] weights
    float* out_idx = (float*)d_out + (size_t)ntok * TOPK;  // [ntok,8] indices (as float)

    const int waves  = ntok / TOKENS_PER_WAVE;             // 1024
    const int blocks = waves / WAVES_PER_BLOCK;            // 256

    router_topk_kernel<<<blocks, WAVES_PER_BLOCK * 32, 0, stream>>>(hidden, Wr, out_w, out_idx);
}